// HyperGCN_20418274525980
// MI455X (gfx1250) — compile-verified
//
#include <hip/hip_runtime.h>
#include <hip/hip_bf16.h>

// ---------------------------------------------------------------------------
// HyperGCN forward on MI455X (gfx1250, wave32).
//   conv(x) = D^-1 H B^-1 H^T (x W) + b
//   out = log_softmax(conv2(relu(conv1(x))))
// GEMMs use V_WMMA_F32_16X16X4_F32 (full fp32 precision, 8-arg builtin).
// Segment sums use native global_atomic_add_f32.
// ---------------------------------------------------------------------------

typedef __attribute__((ext_vector_type(2))) float v2f;
typedef __attribute__((ext_vector_type(8))) float v8f;

#define NN 100000   // nodes
#define NE 50000    // hyperedges
#define NI 3200000  // incidence pairs
#define DIN 128
#define DH 32
#define NC 40

__device__ __forceinline__ void atomic_add_f32(float* p, float v) {
  // result unused -> non-returning global_atomic_add_f32
  __hip_atomic_fetch_add(p, v, __ATOMIC_RELAXED, __HIP_MEMORY_SCOPE_AGENT);
}

__global__ void k_zero(float* __restrict__ p, unsigned n) {
  unsigned i = blockIdx.x * blockDim.x + threadIdx.x;
  if (i < n) p[i] = 0.f;
}

__global__ void k_degrees(const int* __restrict__ nidx, const int* __restrict__ eidx,
                          float* __restrict__ D, float* __restrict__ B, unsigned n) {
  unsigned i = blockIdx.x * blockDim.x + threadIdx.x;
  if (i < n) {
    atomic_add_f32(&D[nidx[i]], 1.f);
    atomic_add_f32(&B[eidx[i]], 1.f);
  }
}

__global__ void k_recip(float* __restrict__ D, float* __restrict__ B) {
  unsigned i = blockIdx.x * blockDim.x + threadIdx.x;
  if (i < NN) { float d = D[i]; D[i] = d > 0.f ? 1.f / d : 0.f; }
  if (i < NE) { float b = B[i]; B[i] = b > 0.f ? 1.f / b : 0.f; }
}

// C[M,N] = A[M,ldA(=K)] * Bm[K,N], fp32 WMMA 16x16x4, row-major everywhere.
// One wave per 16x16 C tile. M assumed multiple of 16; N edge-masked.
// Edge columns are handled branch-free: clamped (in-bounds) load + v_cndmask
// zeroing, so the K loop has no EXEC manipulation (WMMA needs EXEC all-1s).
__global__ void k_gemm_wmma(const float* __restrict__ A, const float* __restrict__ Bm,
                            float* __restrict__ C, int K, int N, int ldA,
                            int mtiles, int ntiles) {
  int tid  = blockIdx.x * blockDim.x + threadIdx.x;
  int w    = tid >> 5;
  if (w >= mtiles * ntiles) return;          // wave-uniform exit -> EXEC all 1s
  int lane = threadIdx.x & 31;
  int tm = w / ntiles, tn = w % ntiles;
  int half = lane >> 4;                      // 0: lanes 0-15, 1: lanes 16-31
  int koff = half * 2;                       // A/B K split per half-wave
  long row = (long)(tm * 16 + (lane & 15));  // A row for this lane
  int  n   = tn * 16 + (lane & 15);          // B/C column for this lane
  bool nv  = (n < N);
  int  nl  = nv ? n : (N - 1);               // clamped, always-in-bounds column
  float msk = nv ? 1.f : 0.f;

  v8f acc = {};
  for (int k0 = 0; k0 < K; k0 += 4) {
    v2f a, b;
    const float* ap = A + row * ldA + k0 + koff;
    a[0] = ap[0];
    a[1] = ap[1];
    // unconditional loads (clamped column), zeroed by select for edge lanes
    b[0] = Bm[(k0 + koff) * N + nl] * msk;
    b[1] = Bm[(k0 + koff + 1) * N + nl] * msk;
    acc = __builtin_amdgcn_wmma_f32_16x16x4_f32(
        /*neg_a=*/false, a, /*neg_b=*/false, b,
        /*c_mod=*/(short)0, acc, /*reuse_a=*/false, /*reuse_b=*/false);
  }

  // C/D layout: VGPR r, lanes 0-15: M=r, lanes 16-31: M=r+8; N = lane&15.
  int m0 = tm * 16 + half * 8;
  if (nv) {
#pragma unroll
    for (int r = 0; r < 8; ++r)
      C[(long)(m0 + r) * N + n] = acc[r];
  }
}

// dst[didx[i]][:] += scale[didx[i]] * src[sidx[i]][:], 4 floats per thread.
__global__ void k_scatter4(const float* __restrict__ src, float* __restrict__ dst,
                           const int* __restrict__ sidx, const int* __restrict__ didx,
                           const float* __restrict__ scale, int F, int chunks,
                           unsigned total) {
  unsigned t = blockIdx.x * blockDim.x + threadIdx.x;
  if (t >= total) return;
  unsigned i = t / chunks;
  int c = (int)(t % chunks) * 4;
  int s = sidx[i], d = didx[i];
  float sc = scale[d];
  float4 v = *(const float4*)(src + (long)s * F + c);
  float* p = dst + (long)d * F + c;
  atomic_add_f32(p + 0, sc * v.x);
  atomic_add_f32(p + 1, sc * v.y);
  atomic_add_f32(p + 2, sc * v.z);
  atomic_add_f32(p + 3, sc * v.w);
}

__global__ void k_bias_relu(float* __restrict__ h, const float* __restrict__ b,
                            unsigned total, int F) {
  unsigned t = blockIdx.x * blockDim.x + threadIdx.x;
  if (t < total) {
    float v = h[t] + b[t % F];
    h[t] = v > 0.f ? v : 0.f;
  }
}

// In-place: row = log_softmax(row + b2). One wave32 per row of 40.
__global__ void k_logsoftmax(float* __restrict__ out, const float* __restrict__ b2) {
  int wid = (int)((blockIdx.x * blockDim.x + threadIdx.x) >> 5);
  if (wid >= NN) return;
  int lane = threadIdx.x & 31;
  float* row = out + (long)wid * NC;
  float z0 = row[lane] + b2[lane];                 // lane < 32 < 40 always valid
  bool v1 = (lane + 32) < NC;                      // lanes 0..7
  float z1 = v1 ? (row[lane + 32] + b2[lane + 32]) : -3.4e38f;
  float m = fmaxf(z0, z1);
#pragma unroll
  for (int off = 16; off > 0; off >>= 1) m = fmaxf(m, __shfl_xor(m, off, 32));
  float s = expf(z0 - m) + (v1 ? expf(z1 - m) : 0.f);
#pragma unroll
  for (int off = 16; off > 0; off >>= 1) s += __shfl_xor(s, off, 32);
  float lse = m + logf(s);
  row[lane] = z0 - lse;
  if (v1) row[lane + 32] = z1 - lse;
}

static inline unsigned nblk(unsigned n, unsigned b) { return (n + b - 1) / b; }

extern "C" void kernel_launch(void* const* d_in, const int* in_sizes, int n_in,
                              void* d_out, int out_size, void* d_ws, size_t ws_size,
                              hipStream_t stream) {
  const float* x        = (const float*)d_in[0];   // [NN, DIN]
  const int*   node_idx = (const int*)d_in[1];     // [NI]
  const int*   edge_idx = (const int*)d_in[2];     // [NI]
  const float* W1       = (const float*)d_in[3];   // [DIN, DH]
  const float* b1       = (const float*)d_in[4];   // [DH]
  const float* W2       = (const float*)d_in[5];   // [DH, NC]
  const float* b2       = (const float*)d_in[6];   // [NC]
  float* out = (float*)d_out;                      // [NN, NC] log-softmax
  float* ws  = (float*)d_ws;

  // Workspace layout (floats); zeroed regions are contiguous at the front.
  float* Dinv = ws;                                // NN
  float* Binv = Dinv + NN;                         // NE
  float* h1   = Binv + NE;                         // NN*DH  (conv1 accumulator -> h)
  float* e1   = h1 + (long)NN * DH;                // NE*DH
  float* xw   = e1 + (long)NE * DH;                // NN*DH
  float* hw2  = xw + (long)NN * DH;                // NN*NC
  float* e2   = hw2 + (long)NN * NC;               // NE*NC
  // total = 14,150,000 floats = 56.6 MB

  // ---- zero accumulators (every launch; graph-replay safe) ----
  unsigned z1 = NN + NE + (unsigned)NN * DH + (unsigned)NE * DH; // Dinv..e1
  k_zero<<<nblk(z1, 256), 256, 0, stream>>>(ws, z1);
  unsigned z2 = (unsigned)NE * NC;
  k_zero<<<nblk(z2, 256), 256, 0, stream>>>(e2, z2);
  unsigned z3 = (unsigned)NN * NC;
  k_zero<<<nblk(z3, 256), 256, 0, stream>>>(out, z3);

  // ---- degrees and inverses ----
  k_degrees<<<nblk(NI, 256), 256, 0, stream>>>(node_idx, edge_idx, Dinv, Binv, NI);
  k_recip<<<nblk(NN, 256), 256, 0, stream>>>(Dinv, Binv);

  // ---- layer 1: xw = x @ W1  (WMMA fp32) ----
  {
    int mt = NN / 16, nt = DH / 16;                // 6250 x 2 tiles
    unsigned thr = (unsigned)mt * nt * 32;
    k_gemm_wmma<<<nblk(thr, 256), 256, 0, stream>>>(x, W1, xw, DIN, DH, DIN, mt, nt);
  }
  // e1[e] += Binv[e] * xw[v] ; h1[v] += Dinv[v] * e1[e]
  k_scatter4<<<nblk(NI * 8u, 256), 256, 0, stream>>>(xw, e1, node_idx, edge_idx, Binv, DH, 8, NI * 8u);
  k_scatter4<<<nblk(NI * 8u, 256), 256, 0, stream>>>(e1, h1, edge_idx, node_idx, Dinv, DH, 8, NI * 8u);
  k_bias_relu<<<nblk((unsigned)NN * DH, 256), 256, 0, stream>>>(h1, b1, (unsigned)NN * DH, DH);

  // ---- layer 2: hw2 = h1 @ W2  (WMMA fp32, N=40 edge-masked) ----
  {
    int mt = NN / 16, nt = (NC + 15) / 16;         // 6250 x 3 tiles
    unsigned thr = (unsigned)mt * nt * 32;
    k_gemm_wmma<<<nblk(thr, 256), 256, 0, stream>>>(h1, W2, hw2, DH, NC, DH, mt, nt);
  }
  k_scatter4<<<nblk(NI * 10u, 256), 256, 0, stream>>>(hw2, e2, node_idx, edge_idx, Binv, NC, 10, NI * 10u);
  k_scatter4<<<nblk(NI * 10u, 256), 256, 0, stream>>>(e2, out, edge_idx, node_idx, Dinv, NC, 10, NI * 10u);

  // ---- + b2 and log_softmax, in place ----
  k_logsoftmax<<<nblk((unsigned)NN * 32, 256), 256, 0, stream>>>(out, b2);
}